// AKConv_88596585382273
// MI455X (gfx1250) — compile-verified
//
#include <hip/hip_runtime.h>
#include <hip/hip_bf16.h>
#include <math.h>

// ---------------------------------------------------------------------------
// AKConv fused pipeline for gfx1250 (MI455X), fp32 end-to-end.
//   Kernel 0: repack cw into WMMA B-fragment order (one float2 per lane/chunk)
//   Kernel 1: offset 3x3 conv -> bilinear indices (int4) + weights (float4)
//   Kernel 2: gather-to-LDS + WMMA f32 16x16x4 GEMM + BN + SiLU epilogue
// ---------------------------------------------------------------------------

#define BATCH 8
#define C1    128
#define C2    256
#define NPAR  5
#define HH    80
#define WW    80
#define HW    (HH * WW)          // 6400
#define KTOT  (C1 * NPAR)        // 640
#define NT    (C2 / 16)          // 16 N-tiles
#define NCHUNK (NPAR * (C1 / 4)) // 160 K-chunks of 4
#define LDA   132                // padded LDS row stride (floats), avoids bank conflicts
#define BN_EPS 1e-5f

typedef __attribute__((ext_vector_type(2))) float v2f;
typedef __attribute__((ext_vector_type(8))) float v8f;

// ---------------------------------------------------------------------------
// Kernel 0: repack cw [C2][C1][n] -> bpack [nt][chunk][lane] (float2), the
// exact per-lane B fragment for V_WMMA_F32_16X16X4_F32:
//   lane: n = lane&15 (column), half = lane>>4; fragment = (K=2*half, 2*half+1)
//   chunk = kk*32 + ch ; K-local j -> c1 = ch*4 + 2*half + j, k = kk*128 + c1.
// ---------------------------------------------------------------------------
__global__ __launch_bounds__(256) void akconv_pack_b(
    const float* __restrict__ cw,   // [C2, C1, n]
    float2* __restrict__ bpack)     // [NT, NCHUNK, 32]
{
    int gid = blockIdx.x * 256 + threadIdx.x;
    if (gid >= NT * NCHUNK * 32) return;

    int lane  = gid & 31;
    int chunk = (gid >> 5) % NCHUNK;
    int nt    = gid / (NCHUNK * 32);

    int n    = lane & 15;
    int half = lane >> 4;
    int kk   = chunk >> 5;       // /32
    int ch   = chunk & 31;
    int c2   = nt * 16 + n;
    int c1a  = ch * 4 + 2 * half;

    float2 v;
    v.x = cw[(size_t)c2 * KTOT + (c1a + 0) * NPAR + kk];
    v.y = cw[(size_t)c2 * KTOT + (c1a + 1) * NPAR + kk];
    bpack[gid] = v;
}

// ---------------------------------------------------------------------------
// Kernel 1: one thread per (b, k, pixel).
// Computes offset_x/offset_y via the 3x3 conv (pad 1), then the four bilinear
// gather indices and weights shared by all 128 input channels.
// ---------------------------------------------------------------------------
__global__ __launch_bounds__(256) void akconv_offsets(
    const float* __restrict__ x,   // [B, C1, H, W]
    const float* __restrict__ pw,  // [2n, C1, 3, 3]
    const float* __restrict__ pb,  // [2n]
    int4*  __restrict__ idx4,      // [B, n, HW]
    float4* __restrict__ g4)       // [B, n, HW]
{
    int gid = blockIdx.x * 256 + threadIdx.x;
    if (gid >= BATCH * NPAR * HW) return;

    int pix = gid % HW;
    int t   = gid / HW;
    int k   = t % NPAR;
    int b   = t / NPAR;
    int ho  = pix / WW;
    int wo  = pix % WW;

    const float* xb = x + (size_t)b * C1 * HW;
    float offx = pb[k];
    float offy = pb[k + NPAR];

    for (int c1 = 0; c1 < C1; ++c1) {
        const float* xc = xb + (size_t)c1 * HW;
        const float* wx = pw + ((size_t)k * C1 + c1) * 9;
        const float* wy = pw + ((size_t)(k + NPAR) * C1 + c1) * 9;
#pragma unroll
        for (int dy = 0; dy < 3; ++dy) {
            int hy = ho + dy - 1;
            if ((unsigned)hy >= (unsigned)HH) continue;
#pragma unroll
            for (int dx = 0; dx < 3; ++dx) {
                int wz = wo + dx - 1;
                if ((unsigned)wz >= (unsigned)WW) continue;
                float xv = xc[hy * WW + wz];
                offx = fmaf(xv, wx[dy * 3 + dx], offx);
                offy = fmaf(xv, wy[dy * 3 + dx], offy);
            }
        }
    }

    // Base kernel sampling offsets for n=5 (matches _get_pn)
    const float pnx[NPAR] = {0.f, 0.f, 1.f, 1.f, 2.f};
    const float pny[NPAR] = {0.f, 1.f, 0.f, 1.f, 0.f};

    float px = offx + (float)ho + pnx[k];
    float py = offy + (float)wo + pny[k];

    float pf = floorf(px);
    float qf = floorf(py);
    float qx_lt = fminf(fmaxf(pf,        0.f), (float)(HH - 1));
    float qy_lt = fminf(fmaxf(qf,        0.f), (float)(WW - 1));
    float qx_rb = fminf(fmaxf(pf + 1.f,  0.f), (float)(HH - 1));
    float qy_rb = fminf(fmaxf(qf + 1.f,  0.f), (float)(WW - 1));
    float pxc   = fminf(fmaxf(px,        0.f), (float)(HH - 1));
    float pyc   = fminf(fmaxf(py,        0.f), (float)(WW - 1));

    float g_lt = (1.f + (qx_lt - pxc)) * (1.f + (qy_lt - pyc));
    float g_rb = (1.f - (qx_rb - pxc)) * (1.f - (qy_rb - pyc));
    float g_lb = (1.f + (qx_lt - pxc)) * (1.f - (qy_rb - pyc));
    float g_rt = (1.f - (qx_rb - pxc)) * (1.f + (qy_lt - pyc));

    int xlt = (int)qx_lt, ylt = (int)qy_lt;
    int xrb = (int)qx_rb, yrb = (int)qy_rb;

    idx4[gid] = make_int4(xlt * WW + ylt, xrb * WW + yrb,
                          xlt * WW + yrb, xrb * WW + ylt);
    g4[gid]   = make_float4(g_lt, g_rb, g_lb, g_rt);
}

// ---------------------------------------------------------------------------
// Kernel 2: one block (8 wave32) per (b, 16-pixel M-tile).
//   Phase 1: cooperatively gather the A tile [5][16 pix][128 c1] into LDS.
//   Phase 2: each wave owns 32 output channels = two 16x16 WMMA tiles and
//            runs 160 K-chunks of v_wmma_f32_16x16x4_f32 with coalesced b64
//            B-fragment loads from bpack, then fused BN+SiLU.
// ---------------------------------------------------------------------------
__global__ __launch_bounds__(256) void akconv_main(
    const float* __restrict__ x,      // [B, C1, HW]
    const float2* __restrict__ bpack, // [NT, NCHUNK, 32]
    const int4*  __restrict__ idx4,   // [B, n, HW]
    const float4* __restrict__ g4,    // [B, n, HW]
    const float* __restrict__ bng,
    const float* __restrict__ bnb,
    const float* __restrict__ bnm,
    const float* __restrict__ bnv,
    float* __restrict__ out)          // [B, C2, HW]
{
    __shared__ float As[NPAR][16][LDA];   // 42,240 B of the 320 KB WGP LDS

    const int tid  = threadIdx.x;
    const int lane = tid & 31;
    const int wave = tid >> 5;

    const int mt   = blockIdx.x % (HW / 16);
    const int b    = blockIdx.x / (HW / 16);
    const int pix0 = mt * 16;

    const float* xb = x + (size_t)b * C1 * HW;

    // ---- Phase 1: build x_off tile in LDS (gathers shared across waves) ----
    for (int e = tid; e < NPAR * 16 * C1; e += 256) {
        int kk  = e / (16 * C1);
        int rem = e - kk * (16 * C1);
        int pix = rem >> 7;       // /128
        int c1  = rem & 127;
        int s   = (b * NPAR + kk) * HW + pix0 + pix;  // broadcast across c1 lanes
        int4   q = idx4[s];
        float4 g = g4[s];
        const float* xc = xb + (size_t)c1 * HW;
        As[kk][pix][c1] = g.x * xc[q.x] + g.y * xc[q.y]
                        + g.z * xc[q.z] + g.w * xc[q.w];
    }
    __syncthreads();

    // ---- Phase 2: WMMA GEMM over K = kk*128 + c1 ----
    const int m_lane = lane & 15;     // A row / C column index
    const int half   = lane >> 4;
    const int kbase  = half * 2;      // local K held by this lane half

    const int c2a = wave * 32 + m_lane;   // N-tile 0 column for this lane
    const int c2b = c2a + 16;             // N-tile 1 column

    // B fragment streams (lane-coalesced float2 per chunk)
    const float2* Bf0 = bpack + ((size_t)(wave * 2)     * NCHUNK) * 32 + lane;
    const float2* Bf1 = bpack + ((size_t)(wave * 2 + 1) * NCHUNK) * 32 + lane;
    __builtin_prefetch(Bf0, 0, 3);               // global_prefetch_b8
    __builtin_prefetch(Bf1, 0, 3);

    v8f acc0 = {};
    v8f acc1 = {};

    for (int kk = 0; kk < NPAR; ++kk) {
        const float* Arow = &As[kk][m_lane][0];
#pragma unroll 4
        for (int ch = 0; ch < C1 / 4; ++ch) {
            int chunk = kk * 32 + ch;
            v2f a;                              // A 16x4 f32 fragment (2 VGPR)
            a.x = Arow[ch * 4 + kbase];         // K = 2*half
            a.y = Arow[ch * 4 + kbase + 1];     // K = 2*half + 1
            float2 f0 = Bf0[(size_t)chunk * 32];   // coalesced global_load_b64
            float2 f1 = Bf1[(size_t)chunk * 32];
            v2f b0; b0.x = f0.x; b0.y = f0.y;
            v2f b1; b1.x = f1.x; b1.y = f1.y;
            acc0 = __builtin_amdgcn_wmma_f32_16x16x4_f32(
                false, a, false, b0, (short)0, acc0, false, false);
            acc1 = __builtin_amdgcn_wmma_f32_16x16x4_f32(
                false, a, false, b1, (short)0, acc1, false, false);
        }
    }

    // ---- Epilogue: BatchNorm(eval) + SiLU, write [B, C2, HW] ----
    float inv0 = bng[c2a] / sqrtf(bnv[c2a] + BN_EPS);
    float inv1 = bng[c2b] / sqrtf(bnv[c2b] + BN_EPS);
    float mu0 = bnm[c2a], be0 = bnb[c2a];
    float mu1 = bnm[c2b], be1 = bnb[c2b];

    float* o0 = out + ((size_t)b * C2 + c2a) * HW + pix0;
    float* o1 = out + ((size_t)b * C2 + c2b) * HW + pix0;

#pragma unroll
    for (int r = 0; r < 8; ++r) {
        int m = r + 8 * half;                 // C/D row for this VGPR
        float v0 = (acc0[r] - mu0) * inv0 + be0;
        float v1 = (acc1[r] - mu1) * inv1 + be1;
        v0 = v0 / (1.f + __expf(-v0));        // SiLU
        v1 = v1 / (1.f + __expf(-v1));
        o0[m] = v0;
        o1[m] = v1;
    }
}

// ---------------------------------------------------------------------------
extern "C" void kernel_launch(void* const* d_in, const int* in_sizes, int n_in,
                              void* d_out, int out_size, void* d_ws, size_t ws_size,
                              hipStream_t stream) {
    const float* x   = (const float*)d_in[0];
    const float* pw  = (const float*)d_in[1];
    const float* pb  = (const float*)d_in[2];
    const float* cw  = (const float*)d_in[3];
    const float* bng = (const float*)d_in[4];
    const float* bnb = (const float*)d_in[5];
    const float* bnm = (const float*)d_in[6];
    const float* bnv = (const float*)d_in[7];
    float* out = (float*)d_out;

    const size_t nSamp = (size_t)BATCH * NPAR * HW;   // 256,000
    int4*   idx4  = (int4*)d_ws;
    float4* g4    = (float4*)((char*)d_ws + nSamp * sizeof(int4));
    float2* bpack = (float2*)((char*)d_ws + nSamp * (sizeof(int4) + sizeof(float4)));
    // ws usage: 4 MB + 4 MB + 640 KB

    int totalB = NT * NCHUNK * 32;                    // 81,920 float2 fragments
    akconv_pack_b<<<(totalB + 255) / 256, 256, 0, stream>>>(cw, bpack);

    int total1 = (int)nSamp;
    akconv_offsets<<<(total1 + 255) / 256, 256, 0, stream>>>(x, pw, pb, idx4, g4);

    int blocks2 = BATCH * (HW / 16);                  // 3,200 blocks
    akconv_main<<<blocks2, 256, 0, stream>>>(x, bpack, idx4, g4,
                                             bng, bnb, bnm, bnv, out);
}